// CPQuadRankLayer_49125835932049
// MI455X (gfx1250) — compile-verified
//
#include <hip/hip_runtime.h>

// CDNA5 / gfx1250: wave32, WMMA (not MFMA).
typedef float v2f __attribute__((ext_vector_type(2)));
typedef float v8f __attribute__((ext_vector_type(8)));

#define B_   16
#define N_   2048
#define IN_  256
#define R_   64
#define OUT_ 256

#define XQ_STRIDE 260   // 256 + 4-bank skew (260 % 64 == 4)
#define P_STRIDE  68    // 64 + skew

__launch_bounds__(128, 1)
__global__ void cpquad_rank_kernel(const float* __restrict__ x,
                                   const float* __restrict__ f_tl,
                                   const float* __restrict__ f_tr,
                                   const float* __restrict__ f_bl,
                                   const float* __restrict__ f_br,
                                   const float* __restrict__ f_out,
                                   const float* __restrict__ gain,
                                   float* __restrict__ out)
{
    __shared__ float xq[B_][XQ_STRIDE];      // staged x for current quadrant
    __shared__ float resid[B_][XQ_STRIDE];   // mean over 4 quadrants
    __shared__ float Pq[B_][P_STRIDE];       // projection result (16 x 64)
    __shared__ float merged[B_][P_STRIDE];   // running product
    __shared__ float invr[B_];               // 1/rms per batch row

    const int n    = blockIdx.x;
    const int tid  = threadIdx.x;
    const int lane = tid & 31;
    const int wave = tid >> 5;          // 4 wave32 per block
    const int ln   = lane & 15;         // row/col index within 16
    const int khi  = lane >> 4;         // K half-select for 16x16x4 fragments

    // ---- init: merged = gain[n], resid = 0 ----------------------------------
    const float g = gain[n];
    for (int i = tid; i < B_ * R_; i += 128)
        merged[i / R_][i % R_] = g;
    for (int i = tid; i < B_ * IN_; i += 128)
        resid[i / IN_][i % IN_] = 0.0f;
    __syncthreads();

    // ---- 4 quadrants, processed sequentially --------------------------------
    for (int q = 0; q < 4; ++q) {
        const float* __restrict__ Fq =
            (q == 0) ? f_tl : (q == 1) ? f_tr : (q == 2) ? f_bl : f_br;
        const float* __restrict__ Fbase = Fq + (long)n * R_ * IN_;

        // stage x[:, n, q, :] into LDS (float4, coalesced) + residual accumulate
        for (int j4 = tid; j4 < (B_ * IN_) / 4; j4 += 128) {
            const int b = (j4 * 4) / IN_;
            const int i = (j4 * 4) % IN_;
            const float4 v = *(const float4*)&x[(((long)b * N_ + n) * 4 + q) * IN_ + i];
            *(float4*)&xq[b][i] = v;
            float4 rv = *(float4*)&resid[b][i];
            rv.x += 0.25f * v.x; rv.y += 0.25f * v.y;
            rv.z += 0.25f * v.z; rv.w += 0.25f * v.w;
            *(float4*)&resid[b][i] = rv;
        }
        __syncthreads();

        // projection: wave w computes r-tile rt = w :  P[16b x 16r], K = 256
        {
            const int rt = wave;
            const float* __restrict__ Brow = &Fbase[(rt * 16 + ln) * IN_];
            v8f c = {0.f, 0.f, 0.f, 0.f, 0.f, 0.f, 0.f, 0.f};
#pragma unroll 4
            for (int k0 = 0; k0 < IN_; k0 += 4) {
                // A (x tile): lane ln = batch row, khi selects K pair {0,1}/{2,3}
                v2f a = *(const v2f*)&xq[ln][k0 + 2 * khi];
                // B (factor^T tile): lane ln = rank col; stream lane's factor row
                v2f b = *(const v2f*)&Brow[k0 + 2 * khi];
                c = __builtin_amdgcn_wmma_f32_16x16x4_f32(
                        false, a, false, b, (short)0, c, false, false);
            }
            // C/D layout: VGPR v -> row (v + 8*khi), col ln
#pragma unroll
            for (int vv = 0; vv < 8; ++vv)
                Pq[vv + 8 * khi][rt * 16 + ln] = c[vv];
        }
        __syncthreads();

        // rms over R=64 per batch row
        if (tid < B_) {
            float s = 0.f;
#pragma unroll 8
            for (int r = 0; r < R_; ++r) { const float p = Pq[tid][r]; s += p * p; }
            invr[tid] = rsqrtf(s * (1.0f / R_) + 1e-6f);
        }
        __syncthreads();

        // merged *= normalized projection
        for (int i = tid; i < B_ * R_; i += 128) {
            const int b = i / R_, r = i % R_;
            merged[b][r] *= Pq[b][r] * invr[b];
        }
        __syncthreads();
    }

    // ---- output GEMM: [16 x 64] @ [64 x 256], + residual --------------------
    const float* __restrict__ Fo = f_out + (long)n * R_ * OUT_;
    __builtin_prefetch(Fo, 0, 1);

    for (int t = 0; t < 4; ++t) {
        const int jt = wave + 4 * t;        // 16 output tiles / 4 waves
        const int j  = jt * 16 + ln;        // lane's output column (coalesced)
        v8f c = {0.f, 0.f, 0.f, 0.f, 0.f, 0.f, 0.f, 0.f};
#pragma unroll 4
        for (int k0 = 0; k0 < R_; k0 += 4) {
            v2f a = *(const v2f*)&merged[ln][k0 + 2 * khi];
            v2f b;
            b.x = Fo[(k0 + 2 * khi)     * OUT_ + j];
            b.y = Fo[(k0 + 2 * khi + 1) * OUT_ + j];
            c = __builtin_amdgcn_wmma_f32_16x16x4_f32(
                    false, a, false, b, (short)0, c, false, false);
        }
#pragma unroll
        for (int vv = 0; vv < 8; ++vv) {
            const int m = vv + 8 * khi;
            out[((long)m * N_ + n) * OUT_ + j] = c[vv] + resid[m][j];
        }
    }
}

extern "C" void kernel_launch(void* const* d_in, const int* in_sizes, int n_in,
                              void* d_out, int out_size, void* d_ws, size_t ws_size,
                              hipStream_t stream) {
    (void)in_sizes; (void)n_in; (void)out_size; (void)d_ws; (void)ws_size;
    const float* x     = (const float*)d_in[0];
    const float* f_tl  = (const float*)d_in[1];
    const float* f_tr  = (const float*)d_in[2];
    const float* f_bl  = (const float*)d_in[3];
    const float* f_br  = (const float*)d_in[4];
    const float* f_out = (const float*)d_in[5];
    const float* gain  = (const float*)d_in[6];
    float* out = (float*)d_out;

    cpquad_rank_kernel<<<N_, 128, 0, stream>>>(x, f_tl, f_tr, f_bl, f_br,
                                               f_out, gain, out);
}